// PretrainRNNModel_4724464026040
// MI455X (gfx1250) — compile-verified
//
#include <hip/hip_runtime.h>
#include <math.h>

typedef __attribute__((ext_vector_type(2))) float v2f;
typedef __attribute__((ext_vector_type(8))) float v8f;

#define T_    100
#define B_    128
#define EMB_  64
#define RDIM_ 32
#define SDIM_ 32
#define IN_   128   // EMB+RDIM+SDIM
#define H_    512
#define ITEMS_ 10000
#define KLIN_ 544   // H + SDIM

__device__ __forceinline__ v2f ldv2(const float* p) { return *(const v2f*)p; }

__device__ __forceinline__ v8f zero8() {
  v8f z = {0.f,0.f,0.f,0.f,0.f,0.f,0.f,0.f};
  return z;
}

// D = A(16x4 f32) * B(4x16 f32) + C, fp32 WMMA (wave32)
__device__ __forceinline__ v8f wmma4(v2f a, v2f b, v8f c) {
  return __builtin_amdgcn_wmma_f32_16x16x4_f32(
      /*neg_a=*/false, a, /*neg_b=*/false, b,
      /*c_mod=*/(short)0, c, /*reuse_a=*/false, /*reuse_b=*/false);
}

__device__ __forceinline__ float sigmoidf_(float x) { return 1.0f / (1.0f + expf(-x)); }

// ---------------------------------------------------------------------------
// Phase 1: gather embedding + one-hot reward bucket + statistic concat.
// Writes ars[T*B, 128] and the statistic columns of lin_in[T*B, 544].
// ---------------------------------------------------------------------------
__global__ void build_inputs(const int* __restrict__ actions,
                             const float* __restrict__ rewards,
                             const float* __restrict__ stats,
                             const float* __restrict__ emb,
                             float* __restrict__ ars,
                             float* __restrict__ lin_in) {
  const int row = blockIdx.x;      // t*B + b
  const int j   = threadIdx.x;     // 0..127
  const int a   = actions[row];
  const float r = rewards[row];
  const int idx = (int)floorf((float)RDIM_ * (2.0f - r) / 4.0f);
  float v;
  if (j < EMB_)                 v = emb[(size_t)a * EMB_ + j];
  else if (j < EMB_ + RDIM_)    v = (j - EMB_ == idx) ? 1.0f : 0.0f;
  else                          v = stats[(size_t)row * SDIM_ + (j - (EMB_ + RDIM_))];
  ars[(size_t)row * IN_ + j] = v;
  if (j >= EMB_ + RDIM_)
    lin_in[(size_t)row * KLIN_ + H_ + (j - (EMB_ + RDIM_))] = v;
}

// ---------------------------------------------------------------------------
// Phase 2: one LSTM timestep.  gates = ars[t] @ W_ih^T + h @ W_hh^T + b.
// One wave owns a 16(batch)x16(hidden) tile for all 4 gates (i,f,g,o), so the
// A operand is loaded once per k-step and shared by 4 WMMAs; the nonlinearity
// + cell update runs in the C/D register layout with no shuffles.
// Grid: 32 blocks x 256 threads = 256 waves = 8 row-tiles * 32 col-tiles.
// ---------------------------------------------------------------------------
__global__ void __launch_bounds__(256) lstm_step(
    int t,
    const float* __restrict__ ars,
    const float* __restrict__ W_ih,   // [2048,128]
    const float* __restrict__ W_hh,   // [2048,512]
    const float* __restrict__ b_ih,   // [2048]
    const float* __restrict__ b_hh,   // [2048]
    const float* __restrict__ h_prev, // [128,512]
    const float* __restrict__ c_prev, // [128,512]
    float* __restrict__ h_out,
    float* __restrict__ c_out,
    float* __restrict__ lin_in) {
  const int lane = threadIdx.x & 31;
  const int wid  = threadIdx.x >> 5;
  const int wave = blockIdx.x * 8 + wid;  // 0..255
  const int rt   = wave >> 5;             // 0..7  (batch tile)
  const int ct   = wave & 31;             // 0..31 (hidden tile)
  const int row0 = rt * 16;
  const int col0 = ct * 16;
  const int m    = lane & 15;
  const int n    = lane & 15;
  const int half = lane >> 4;

  v8f acc[4];
#pragma unroll
  for (int g = 0; g < 4; ++g) acc[g] = zero8();

  // h_prev @ W_hh^T  (K = 512)
  const float* hA = h_prev + (size_t)(row0 + m) * H_ + 2 * half;
  for (int k = 0; k < H_; k += 4) {
    v2f a = ldv2(hA + k);
#pragma unroll
    for (int g = 0; g < 4; ++g) {
      v2f b = ldv2(W_hh + (size_t)(g * H_ + col0 + n) * H_ + k + 2 * half);
      acc[g] = wmma4(a, b, acc[g]);
    }
  }
  // ars[t] @ W_ih^T  (K = 128), fused input projection
  const float* xA = ars + ((size_t)t * B_ + row0 + m) * IN_ + 2 * half;
  for (int k = 0; k < IN_; k += 4) {
    v2f a = ldv2(xA + k);
#pragma unroll
    for (int g = 0; g < 4; ++g) {
      v2f b = ldv2(W_ih + (size_t)(g * H_ + col0 + n) * IN_ + k + 2 * half);
      acc[g] = wmma4(a, b, acc[g]);
    }
  }

  const int col = col0 + n;
  float bias[4];
#pragma unroll
  for (int g = 0; g < 4; ++g) bias[g] = b_ih[g * H_ + col] + b_hh[g * H_ + col];

#pragma unroll
  for (int r = 0; r < 8; ++r) {
    const int row = row0 + r + 8 * half;  // batch index
    const float gi = sigmoidf_(acc[0][r] + bias[0]);
    const float gf = sigmoidf_(acc[1][r] + bias[1]);
    const float gg = tanhf(acc[2][r] + bias[2]);
    const float go = sigmoidf_(acc[3][r] + bias[3]);
    const float c  = gf * c_prev[(size_t)row * H_ + col] + gi * gg;
    const float h  = go * tanhf(c);
    c_out[(size_t)row * H_ + col] = c;
    h_out[(size_t)row * H_ + col] = h;
    lin_in[((size_t)t * B_ + row) * KLIN_ + col] = h;
  }
}

// ---------------------------------------------------------------------------
// Phase 3: output projection [12800,544] @ [544,10000]^T + b_lin.
// One wave computes a 16x80 strip (5 tiles share every A load).
// Grid: 12500 blocks x 256 threads = 100000 waves = 800 row-tiles * 125 strips.
// ---------------------------------------------------------------------------
__global__ void __launch_bounds__(256) out_gemm(
    const float* __restrict__ lin_in,  // [12800,544]
    const float* __restrict__ W_lin,   // [10000,544]
    const float* __restrict__ b_lin,   // [10000]
    float* __restrict__ out) {         // [12800,10000]
  const int lane = threadIdx.x & 31;
  const int wid  = threadIdx.x >> 5;
  const int wave = blockIdx.x * 8 + wid;  // 0..99999
  const int rt   = wave / 125;            // 0..799
  const int s    = wave % 125;            // 0..124
  const int row0 = rt * 16;
  const int col0 = s * 80;
  const int m    = lane & 15;
  const int n    = lane & 15;
  const int half = lane >> 4;

  v8f acc[5];
#pragma unroll
  for (int j = 0; j < 5; ++j) acc[j] = zero8();

  const float* A  = lin_in + (size_t)(row0 + m) * KLIN_ + 2 * half;
  const float* Bp = W_lin + 2 * half;
  __builtin_prefetch(A, 0, 1);  // global_prefetch_b8
  for (int k = 0; k < KLIN_; k += 4) {
    v2f a = ldv2(A + k);
#pragma unroll
    for (int j = 0; j < 5; ++j) {
      v2f b = ldv2(Bp + (size_t)(col0 + j * 16 + n) * KLIN_ + k);
      acc[j] = wmma4(a, b, acc[j]);
    }
  }

#pragma unroll
  for (int j = 0; j < 5; ++j) {
    const int col = col0 + j * 16 + n;
    const float bl = b_lin[col];
#pragma unroll
    for (int r = 0; r < 8; ++r) {
      const int row = row0 + r + 8 * half;
      out[(size_t)row * ITEMS_ + col] = acc[j][r] + bl;
    }
  }
}

// ---------------------------------------------------------------------------
// Norms: deterministic two-stage sum-of-squares reduction.
// ---------------------------------------------------------------------------
__global__ void sumsq_partial(const float* __restrict__ x, long long nelem,
                              float* __restrict__ partials) {
  __shared__ float sm[256];
  float s = 0.f;
  for (long long i = (long long)blockIdx.x * 256 + threadIdx.x; i < nelem;
       i += (long long)gridDim.x * 256) {
    float v = x[i];
    s += v * v;
  }
  sm[threadIdx.x] = s;
  __syncthreads();
  for (int off = 128; off > 0; off >>= 1) {
    if (threadIdx.x < off) sm[threadIdx.x] += sm[threadIdx.x + off];
    __syncthreads();
  }
  if (threadIdx.x == 0) partials[blockIdx.x] = sm[0];
}

__global__ void norm_finalize(const float* __restrict__ partialsW, int npw,
                              const float* __restrict__ b_lin,
                              float* __restrict__ out_scalar) {
  __shared__ float sm[256];
  float s = 0.f;
  for (int i = threadIdx.x; i < ITEMS_; i += 256) {
    float v = b_lin[i];
    s += v * v;
  }
  sm[threadIdx.x] = s;
  __syncthreads();
  for (int off = 128; off > 0; off >>= 1) {
    if (threadIdx.x < off) sm[threadIdx.x] += sm[threadIdx.x + off];
    __syncthreads();
  }
  if (threadIdx.x == 0) {
    float sw = 0.f;
    for (int i = 0; i < npw; ++i) sw += partialsW[i];
    out_scalar[0] = sqrtf(sw) + sqrtf(sm[0]);
  }
}

// ---------------------------------------------------------------------------
extern "C" void kernel_launch(void* const* d_in, const int* in_sizes, int n_in,
                              void* d_out, int out_size, void* d_ws, size_t ws_size,
                              hipStream_t stream) {
  (void)in_sizes; (void)n_in; (void)out_size; (void)ws_size;
  const int*   actions = (const int*)  d_in[0];
  const float* rewards = (const float*)d_in[1];
  const float* stats   = (const float*)d_in[2];
  const float* hx      = (const float*)d_in[3];
  const float* cx      = (const float*)d_in[4];
  const float* emb     = (const float*)d_in[5];
  const float* W_ih    = (const float*)d_in[6];
  const float* W_hh    = (const float*)d_in[7];
  const float* b_ih    = (const float*)d_in[8];
  const float* b_hh    = (const float*)d_in[9];
  const float* W_lin   = (const float*)d_in[10];
  const float* b_lin   = (const float*)d_in[11];
  float* out = (float*)d_out;

  // Workspace layout (~36 MB)
  float* ars      = (float*)d_ws;                       // 12800*128
  float* lin_in   = ars + (size_t)T_ * B_ * IN_;        // 12800*544
  float* h0       = lin_in + (size_t)T_ * B_ * KLIN_;   // 128*512
  float* h1       = h0 + (size_t)B_ * H_;
  float* c0       = h1 + (size_t)B_ * H_;
  float* c1       = c0 + (size_t)B_ * H_;
  float* partials = c1 + (size_t)B_ * H_;               // 512
  const int NPW = 512;

  hipMemcpyAsync(h0, hx, (size_t)B_ * H_ * sizeof(float), hipMemcpyDeviceToDevice, stream);
  hipMemcpyAsync(c0, cx, (size_t)B_ * H_ * sizeof(float), hipMemcpyDeviceToDevice, stream);

  build_inputs<<<T_ * B_, IN_, 0, stream>>>(actions, rewards, stats, emb, ars, lin_in);

  float* hp = h0; float* hn = h1;
  float* cp = c0; float* cn = c1;
  for (int t = 0; t < T_; ++t) {
    lstm_step<<<32, 256, 0, stream>>>(t, ars, W_ih, W_hh, b_ih, b_hh,
                                      hp, cp, hn, cn, lin_in);
    float* tmp;
    tmp = hp; hp = hn; hn = tmp;
    tmp = cp; cp = cn; cn = tmp;
  }

  out_gemm<<<12500, 256, 0, stream>>>(lin_in, W_lin, b_lin, out);

  sumsq_partial<<<NPW, 256, 0, stream>>>(W_lin, (long long)ITEMS_ * KLIN_, partials);
  norm_finalize<<<1, 256, 0, stream>>>(partials, NPW, b_lin,
                                       out + (size_t)T_ * B_ * ITEMS_);
}